// DigitCaps_61349312856200
// MI455X (gfx1250) — compile-verified
//
#include <hip/hip_runtime.h>

// DigitCaps dynamic routing for MI455X (gfx1250, wave32).
// Strategy: materialize u_hat (188.7MB, fits MI455X 192MB L2) once via
// V_WMMA_F32_16X16X4_F32 tiles; routing iterations then stream it from L2.

typedef __attribute__((ext_vector_type(2))) float v2f;
typedef __attribute__((ext_vector_type(8))) float v8f;

#define B_   256
#define R_   1152
#define C_   10
#define O_   16
#define I_   8
#define CO_  160           // C*O
#define NUM_IT 3
#define XPAD (I_ + 1)      // LDS row pad to avoid bank conflicts
#define BRSTRIDE (R_ * CO_)  // u_hat stride between batch rows (floats)

// ---------------------------------------------------------------------------
// Kernel 1: u_hat[b,r,co] = sum_i x[b,r,i] * W[r,co,i]
// One workgroup (8 waves) per r. Per-r GEMM: A=(256x8) x B^T=(8x160).
// 16x16 output tiles, K=8 = 2 x V_WMMA_F32_16X16X4_F32.
// D stores use one base address + compile-time strides so they fold into
// global_store_b32 immediate offsets (stride 737280B < 2^23).
// ---------------------------------------------------------------------------
__global__ __launch_bounds__(256) void uhat_wmma_kernel(
    const float* __restrict__ x,      // (B, R, I)
    const float* __restrict__ W,      // (R, C, O, I)
    float* __restrict__ u_hat) {      // (B, R, CO)
  __shared__ float xs[B_ * XPAD];     // x[:, r, :]  (padded)
  __shared__ float ws[CO_ * XPAD];    // W[r, :, :]  (padded)

  const int r   = blockIdx.x;
  const int tid = threadIdx.x;

  // Stage x[:, r, :] : thread t loads batch-row t (8 contiguous floats).
  {
    const float* xr = x + (size_t)tid * (R_ * I_) + (size_t)r * I_;
#pragma unroll
    for (int j = 0; j < I_; ++j) xs[tid * XPAD + j] = xr[j];
  }
  // Stage W[r] : 160 rows of 8 contiguous floats.
  if (tid < CO_) {
    const float* wr = W + (size_t)r * (CO_ * I_) + tid * I_;
#pragma unroll
    for (int j = 0; j < I_; ++j) ws[tid * XPAD + j] = wr[j];
  }
  __syncthreads();

  const int wv   = tid >> 5;          // wave id 0..7
  const int lane = tid & 31;
  const int m    = lane & 15;         // A: row M / B: col N (per ISA layout)
  const int kk   = (lane >> 4) << 1;  // K sub-pair: lanes 0-15 -> {0,1}, 16-31 -> {2,3}
  const int mrow = (wv * 2) * 16 + ((lane >> 4) << 3);  // first D row this lane writes

  // 160 tiles = 10 n-tiles x 16 m-tiles; wave wv owns m-tiles {2wv, 2wv+1}.
  for (int nt = 0; nt < C_; ++nt) {
    // B fragments (4x16, K-major pair per lane): B[k][n] = W[r, n, k]
    const int co = nt * 16 + m;
    v2f b0, b1;
    b0.x = ws[co * XPAD + kk + 0];
    b0.y = ws[co * XPAD + kk + 1];
    b1.x = ws[co * XPAD + kk + 4];
    b1.y = ws[co * XPAD + kk + 5];

    // Single 64-bit base per (wave, nt); all 16 stores use constant offsets.
    float* dst = u_hat + ((size_t)mrow * R_ + r) * CO_ + co;

#pragma unroll
    for (int mh = 0; mh < 2; ++mh) {
      const int brow = (wv * 2 + mh) * 16 + m;
      v2f a0, a1;                     // A[m][k] = x[brow, r, k]
      a0.x = xs[brow * XPAD + kk + 0];
      a0.y = xs[brow * XPAD + kk + 1];
      a1.x = xs[brow * XPAD + kk + 4];
      a1.y = xs[brow * XPAD + kk + 5];

      v8f acc = {};
      acc = __builtin_amdgcn_wmma_f32_16x16x4_f32(
          false, a0, false, b0, (short)0, acc, false, false);
      acc = __builtin_amdgcn_wmma_f32_16x16x4_f32(
          false, a1, false, b1, (short)0, acc, false, false);

      // D layout: vgpr v, lane L -> M = v + 8*(L>=16), N = L&15.
#pragma unroll
      for (int v = 0; v < 8; ++v)
        dst[(mh * 16 + v) * BRSTRIDE] = acc[v];   // constant byte offsets
    }
  }
}

// ---------------------------------------------------------------------------
// Kernel 2: c_ij[:, c] = softmax over the R axis of b_ij[:, c]. Block per c.
// ---------------------------------------------------------------------------
__global__ __launch_bounds__(256) void softmax_kernel(
    const float* __restrict__ b_ij, float* __restrict__ c_ij) {
  const int c   = blockIdx.x;
  const int tid = threadIdx.x;
  __shared__ float red[256];

  float mx = -__builtin_inff();
  for (int r = tid; r < R_; r += 256) mx = fmaxf(mx, b_ij[r * C_ + c]);
  red[tid] = mx;
  __syncthreads();
  for (int s = 128; s > 0; s >>= 1) {
    if (tid < s) red[tid] = fmaxf(red[tid], red[tid + s]);
    __syncthreads();
  }
  mx = red[0];
  __syncthreads();

  float sum = 0.f;
  for (int r = tid; r < R_; r += 256) sum += __expf(b_ij[r * C_ + c] - mx);
  red[tid] = sum;
  __syncthreads();
  for (int s = 128; s > 0; s >>= 1) {
    if (tid < s) red[tid] += red[tid + s];
    __syncthreads();
  }
  const float inv = 1.f / red[0];
  for (int r = tid; r < R_; r += 256)
    c_ij[r * C_ + c] = __expf(b_ij[r * C_ + c] - mx) * inv;
}

// ---------------------------------------------------------------------------
// Kernel 3: s[b,c,o] = sum_r c_ij[r,c]*u_hat[b,r,c,o]; v = squash(s).
// Block per b (160 threads = 5 waves); lane-contiguous 'co' gives coalesced
// 128B u_hat reads per wave per r. u_hat stays L2-resident on MI455X.
// ---------------------------------------------------------------------------
__global__ __launch_bounds__(160) void sv_kernel(
    const float* __restrict__ u_hat, const float* __restrict__ c_ij,
    float* __restrict__ v_out) {
  const int b  = blockIdx.x;
  const int co = threadIdx.x;        // 0..159
  const int c  = co >> 4;
  const float* up = u_hat + (size_t)b * BRSTRIDE + co;

  float s = 0.f;
  for (int r = 0; r < R_; ++r) {
    __builtin_prefetch(up + (size_t)(r + 16) * CO_, 0, 0);  // global_prefetch_b8
    s = fmaf(c_ij[r * C_ + c], up[(size_t)r * CO_], s);
  }
  const float sq = s * s;
  v_out[b * CO_ + co] = sq * s / ((1.f + sq) * __builtin_sqrtf(sq));
}

// ---------------------------------------------------------------------------
// Kernel 4: b_ij[r,c] += mean_b( sum_o u_hat[b,r,c,o] * v[b,c,o] ).
// One wave32 per (r,c): lanes = (o in 0..15) x (b-parity), shfl_xor reduce.
// ---------------------------------------------------------------------------
__global__ __launch_bounds__(256) void agree_kernel(
    const float* __restrict__ u_hat, const float* __restrict__ v,
    float* __restrict__ b_ij) {
  const int gw   = blockIdx.x * 8 + (threadIdx.x >> 5);  // 0..11519
  const int lane = threadIdx.x & 31;
  const int r = gw / C_;
  const int c = gw % C_;
  const int o  = lane & 15;
  const int bh = lane >> 4;

  const float* up = u_hat + (size_t)r * CO_ + c * O_ + o;
  const float* vp = v + c * O_ + o;

  float acc = 0.f;
  for (int b = bh; b < B_; b += 2)
    acc = fmaf(up[(size_t)b * BRSTRIDE], vp[b * CO_], acc);

#pragma unroll
  for (int off = 16; off > 0; off >>= 1)
    acc += __shfl_xor(acc, off, 32);

  if (lane == 0) b_ij[r * C_ + c] += acc * (1.f / (float)B_);
}

__global__ void zero_kernel(float* __restrict__ p, int n) {
  const int i = blockIdx.x * blockDim.x + threadIdx.x;
  if (i < n) p[i] = 0.f;
}

// ---------------------------------------------------------------------------
extern "C" void kernel_launch(void* const* d_in, const int* in_sizes, int n_in,
                              void* d_out, int out_size, void* d_ws, size_t ws_size,
                              hipStream_t stream) {
  const float* x = (const float*)d_in[0];   // (B, R, I)
  const float* W = (const float*)d_in[1];   // (R, C, O, I)
  float* out = (float*)d_out;               // (B, C, O, 1) == (B, CO) flat

  char* ws = (char*)d_ws;
  float* u_hat = (float*)ws;                                    // 188.7 MB
  size_t off = (size_t)B_ * R_ * CO_ * sizeof(float);
  float* b_ij = (float*)(ws + off); off += (size_t)R_ * C_ * sizeof(float);
  float* c_ij = (float*)(ws + off); off += (size_t)R_ * C_ * sizeof(float);
  float* vbuf = (float*)(ws + off); off += (size_t)B_ * CO_ * sizeof(float);

  zero_kernel<<<(R_ * C_ + 255) / 256, 256, 0, stream>>>(b_ij, R_ * C_);
  uhat_wmma_kernel<<<R_, 256, 0, stream>>>(x, W, u_hat);

  for (int it = 0; it < NUM_IT; ++it) {
    softmax_kernel<<<C_, 256, 0, stream>>>(b_ij, c_ij);
    float* vtgt = (it == NUM_IT - 1) ? out : vbuf;
    sv_kernel<<<B_, 160, 0, stream>>>(u_hat, c_ij, vtgt);
    if (it < NUM_IT - 1)
      agree_kernel<<<(R_ * C_) / 8, 256, 0, stream>>>(u_hat, vbuf, b_ij);
  }
}